// GCN_38852274160023
// MI455X (gfx1250) — compile-verified
//
#include <hip/hip_runtime.h>

typedef __attribute__((ext_vector_type(2))) float v2f;
typedef __attribute__((ext_vector_type(8))) float v8f;

#define DFEAT 64

// ---------------------------------------------------------------- utilities

__global__ void gcn_zero_f32(float* __restrict__ p, int n) {
    int i = blockIdx.x * blockDim.x + threadIdx.x;
    if (i < n) p[i] = 0.0f;
}

// deg[col[e]] += 1 for every edge (degree of A + I counted at targets)
__global__ void gcn_degree(const int* __restrict__ col, float* __restrict__ deg, int E) {
    int e = blockIdx.x * blockDim.x + threadIdx.x;
    if (e < E) atomicAdd(&deg[col[e]], 1.0f);
}

// dinv[i] = rsqrt(deg[i] + 1)   (the +1 is the self loop)
__global__ void gcn_dinv(float* __restrict__ deg, int N) {
    int i = blockIdx.x * blockDim.x + threadIdx.x;
    if (i < N) deg[i] = rsqrtf(deg[i] + 1.0f);
}

// ------------------------------------------------------------- WMMA GEMM
// H[N x 64] = X[N x 64] @ W[64 x 64], fp32 via V_WMMA_F32_16X16X4_F32.
// Block = 256 threads = 8 waves. Wave w computes output tile:
//   rows  [blockIdx.x*32 + (w>>2)*16 , +16)
//   cols  [(w&3)*16 , +16)
// K loop: 16 steps of K=4.
__global__ void gcn_gemm_wmma(const float* __restrict__ X,
                              const float* __restrict__ W,
                              float* __restrict__ H, int nrows) {
    const int lane = threadIdx.x & 31;
    const int wid  = threadIdx.x >> 5;

    const int rowTile = blockIdx.x * 2 + (wid >> 2);
    const int n0      = (wid & 3) * 16;
    const int row0    = rowTile * 16;
    if (row0 >= nrows) return;                 // wave-uniform: EXEC stays all-1s

    const int m  = lane & 15;                  // M (for A/B/C lanes)
    const int kg = (lane >> 4) * 2;            // K sub-offset: 0 or 2

    // clamp row for loads if tile overhangs (stores predicated below)
    int mrow = row0 + m;
    if (mrow >= nrows) mrow = nrows - 1;
    const float* __restrict__ xrow = X + (size_t)mrow * DFEAT;

    v8f c = {};
#pragma unroll
    for (int k0 = 0; k0 < DFEAT; k0 += 4) {
        v2f a, b;
        // A 16x4: lane<16 holds K = k0+{0,1}; lane>=16 holds K = k0+{2,3}
        a.x = xrow[k0 + kg + 0];
        a.y = xrow[k0 + kg + 1];
        // B 4x16: same K striping, N = n0 + m ; W is [K][N] row-major
        b.x = W[(size_t)(k0 + kg + 0) * DFEAT + n0 + m];
        b.y = W[(size_t)(k0 + kg + 1) * DFEAT + n0 + m];
        c = __builtin_amdgcn_wmma_f32_16x16x4_f32(
                /*neg_a=*/false, a, /*neg_b=*/false, b,
                /*c_mod=*/(short)0, c, /*reuse_a=*/false, /*reuse_b=*/false);
    }

    // C/D layout: VGPR j -> M = j + 8*(lane>>4), N = lane&15
    const int mBase = row0 + ((lane >> 4) << 3);
#pragma unroll
    for (int j = 0; j < 8; ++j) {
        int r = mBase + j;
        if (r < nrows) H[(size_t)r * DFEAT + n0 + m] = c[j];
    }
}

// --------------------------------------------------------------- scatter
// One wave per edge; lane k handles features {2k, 2k+1} via float2 gather,
// then two L2-resident f32 atomics into agg[col].
__global__ void gcn_scatter(const int* __restrict__ row,
                            const int* __restrict__ col,
                            const float* __restrict__ dinv,
                            const float* __restrict__ h,
                            float* __restrict__ agg, int E) {
    const int lane = threadIdx.x & 31;
    const int wid  = threadIdx.x >> 5;
    const int e    = blockIdx.x * (blockDim.x >> 5) + wid;
    if (e >= E) return;

    const int r = row[e];
    const int c = col[e];
    const float norm = dinv[r] * dinv[c];

    const float* __restrict__ hr = h + (size_t)r * DFEAT;
    float* __restrict__ ac = agg + (size_t)c * DFEAT;

    const int k = lane * 2;
    const float2 v = *(const float2*)(hr + k);
    atomicAdd(ac + k,     v.x * norm);
    atomicAdd(ac + k + 1, v.y * norm);
}

// -------------------------------------------------------------- finalize
// out = agg + dinv^2 * h + b  (optional ReLU), float4-vectorized (N*16 threads)
__global__ void gcn_finalize(const float* __restrict__ agg,
                             const float* __restrict__ h,
                             const float* __restrict__ dinv,
                             const float* __restrict__ b,
                             float* __restrict__ out, int N, int relu) {
    const int idx = blockIdx.x * blockDim.x + threadIdx.x;
    if (idx >= N * (DFEAT / 4)) return;
    const int i = idx >> 4;             // node
    const int k = (idx & 15) << 2;      // feature base

    float s = dinv[i];
    s = s * s;

    const float4 av = *(const float4*)(agg + (size_t)idx * 4);
    const float4 hv = *(const float4*)(h   + (size_t)idx * 4);
    const float4 bv = *(const float4*)(b + k);

    float4 o;
    o.x = fmaf(s, hv.x, av.x) + bv.x;
    o.y = fmaf(s, hv.y, av.y) + bv.y;
    o.z = fmaf(s, hv.z, av.z) + bv.z;
    o.w = fmaf(s, hv.w, av.w) + bv.w;
    if (relu) {
        o.x = fmaxf(o.x, 0.0f);
        o.y = fmaxf(o.y, 0.0f);
        o.z = fmaxf(o.z, 0.0f);
        o.w = fmaxf(o.w, 0.0f);
    }
    *(float4*)(out + (size_t)idx * 4) = o;
}

// ---------------------------------------------------------------- launch

extern "C" void kernel_launch(void* const* d_in, const int* in_sizes, int n_in,
                              void* d_out, int out_size, void* d_ws, size_t ws_size,
                              hipStream_t stream) {
    const float* x  = (const float*)d_in[0];
    const int*   ei = (const int*)  d_in[1];
    const float* W1 = (const float*)d_in[2];
    const float* b1 = (const float*)d_in[3];
    const float* W2 = (const float*)d_in[4];
    const float* b2 = (const float*)d_in[5];
    const float* W3 = (const float*)d_in[6];
    const float* b3 = (const float*)d_in[7];

    const int N = in_sizes[0] / DFEAT;
    const int E = in_sizes[1] / 2;
    const int* row = ei;
    const int* col = ei + E;

    float* ws   = (float*)d_ws;
    float* dinv = ws;                         // N floats
    float* h    = dinv + N;                   // N*64 floats
    float* agg  = h + (size_t)N * DFEAT;      // N*64 floats
    float* y    = agg + (size_t)N * DFEAT;    // N*64 floats
    float* out  = (float*)d_out;

    const int nf       = N * DFEAT;
    const int gZeroN   = (N + 255) / 256;
    const int gZeroF   = (nf + 255) / 256;
    const int gEdge    = (E + 255) / 256;
    const int gScat    = (E + 7) / 8;                 // 8 waves per block
    const int gGemm    = (N + 31) / 32;               // 32 rows per block
    const int gFin     = (N * (DFEAT / 4) + 255) / 256;

    // --- degree / dinv (fresh each call; ws is not re-zeroed by harness)
    gcn_zero_f32<<<gZeroN, 256, 0, stream>>>(dinv, N);
    gcn_degree  <<<gEdge,  256, 0, stream>>>(col, dinv, E);
    gcn_dinv    <<<gZeroN, 256, 0, stream>>>(dinv, N);

    // --- layer 1: x -> y
    gcn_gemm_wmma<<<gGemm, 256, 0, stream>>>(x, W1, h, N);
    gcn_zero_f32 <<<gZeroF, 256, 0, stream>>>(agg, nf);
    gcn_scatter  <<<gScat, 256, 0, stream>>>(row, col, dinv, h, agg, E);
    gcn_finalize <<<gFin,  256, 0, stream>>>(agg, h, dinv, b1, y, N, 1);

    // --- layer 2: y -> y (in-place safe: finalize reads only agg/h/dinv)
    gcn_gemm_wmma<<<gGemm, 256, 0, stream>>>(y, W2, h, N);
    gcn_zero_f32 <<<gZeroF, 256, 0, stream>>>(agg, nf);
    gcn_scatter  <<<gScat, 256, 0, stream>>>(row, col, dinv, h, agg, E);
    gcn_finalize <<<gFin,  256, 0, stream>>>(agg, h, dinv, b2, y, N, 1);

    // --- layer 3: y -> out (no ReLU)
    gcn_gemm_wmma<<<gGemm, 256, 0, stream>>>(y, W3, h, N);
    gcn_zero_f32 <<<gZeroF, 256, 0, stream>>>(agg, nf);
    gcn_scatter  <<<gScat, 256, 0, stream>>>(row, col, dinv, h, agg, E);
    gcn_finalize <<<gFin,  256, 0, stream>>>(agg, h, dinv, b3, out, N, 0);
}